// QuantLayer_62500364091969
// MI455X (gfx1250) — compile-verified
//
#include <hip/hip_runtime.h>

typedef float v2f __attribute__((ext_vector_type(2)));
typedef float v4f __attribute__((ext_vector_type(4)));
typedef float v8f __attribute__((ext_vector_type(8)));

#define D           64        // vector dim
#define K           1024      // codebook entries
#define ROW_PAD     65        // LDS codebook row stride (64 + 1 anti-bank-conflict pad)
#define WAVES       8
#define BLOCK       (WAVES * 32)
#define ROWS_PER_WAVE 32      // two 16-row M tiles per wave (B fragment reuse x2)
#define ROWS_PER_WG (WAVES * ROWS_PER_WAVE)   // 256

// score(n) = c_sq[n] - 2*dot(x,c_n)  (x^2 is row-constant -> argmin equivalent)
// A pre-scaled by -2; accumulator initialized with c_sq -> WMMA emits scores directly.
extern "C" __global__ __launch_bounds__(BLOCK) void vq_wmma_f32_kernel(
    const float* __restrict__ x,
    const float* __restrict__ cb,
    float* __restrict__ out)
{
    extern __shared__ float smem[];            // [K*ROW_PAD] codebook + [K] c_sq (~270 KB)
    float* s_cb  = smem;
    float* s_csq = smem + K * ROW_PAD;

    const int tid  = threadIdx.x;
    const int lane = tid & 31;
    const int wave = tid >> 5;

    // ---- stage codebook into LDS (padded rows), computing c_sq on the fly ----
    // 256 threads x 4 rows = 1024 rows
    #pragma unroll
    for (int r = 0; r < 4; ++r) {
        const int row = tid * 4 + r;
        const float* src = cb + row * D;
        float* dst = s_cb + row * ROW_PAD;
        float sq = 0.0f;
        #pragma unroll
        for (int j = 0; j < D / 4; ++j) {
            v4f v = *(const v4f*)(src + j * 4);
            *(v4f*)(dst + j * 4) = v;
            sq += v.x * v.x + v.y * v.y + v.z * v.z + v.w * v.w;
        }
        s_csq[row] = sq;
    }
    __syncthreads();

    // ---- load this wave's two A tiles (32 rows of x), scaled by -2 ----
    // A 16x4 f32 layout: lanes 0-15 -> M=lane, K={0,1}; lanes 16-31 -> M=lane-16, K={2,3}
    const int row0 = blockIdx.x * ROWS_PER_WG + wave * ROWS_PER_WAVE;
    const int m    = lane & 15;          // row within M tile / column within B tile
    const int kk   = (lane >> 4) * 2;    // K sub-offset (0 or 2)

    v2f a0[16], a1[16];
    #pragma unroll
    for (int ks = 0; ks < 16; ++ks) {
        v2f v = *(const v2f*)(x + (row0 + m) * D + ks * 4 + kk);
        a0[ks].x = -2.0f * v.x;
        a0[ks].y = -2.0f * v.y;
        v2f w = *(const v2f*)(x + (row0 + 16 + m) * D + ks * 4 + kk);
        a1[ks].x = -2.0f * w.x;
        a1[ks].y = -2.0f * w.y;
    }

    // ---- sweep codebook tiles; B fragments AND c_sq register double-buffered ----
    float best0[8], best1[8];
    int   bidx0[8], bidx1[8];
    #pragma unroll
    for (int r = 0; r < 8; ++r) {
        best0[r] = 3.4e38f; bidx0[r] = 0;
        best1[r] = 3.4e38f; bidx1[r] = 0;
    }

    v2f   bc[16];                     // current tile's B fragments
    float csc;                        // current tile's c_sq (this lane's column)
    #pragma unroll
    for (int ks = 0; ks < 16; ++ks)
        bc[ks] = *(const v2f*)(s_cb + m * ROW_PAD + ks * 4 + kk);
    csc = s_csq[m];

    for (int t = 0; t < K / 16; ++t) {
        const int n0 = t * 16;
        const int n1 = ((t + 1) & (K / 16 - 1)) * 16;   // prefetch target (wraps)

        // prefetch next tile's B fragments + c_sq while this tile computes
        v2f bn[16];
        #pragma unroll
        for (int ks = 0; ks < 16; ++ks)
            bn[ks] = *(const v2f*)(s_cb + (n1 + m) * ROW_PAD + ks * 4 + kk);
        const float csn = s_csq[n1 + m];

        v8f acc0, acc1;
        #pragma unroll
        for (int r = 0; r < 8; ++r) { acc0[r] = csc; acc1[r] = csc; }

        // two independent accumulator chains share each B fragment
        #pragma unroll
        for (int ks = 0; ks < 16; ++ks) {
            acc0 = __builtin_amdgcn_wmma_f32_16x16x4_f32(
                false, a0[ks], false, bc[ks], (short)0, acc0, false, false);
            acc1 = __builtin_amdgcn_wmma_f32_16x16x4_f32(
                false, a1[ks], false, bc[ks], (short)0, acc1, false, false);
        }

        #pragma unroll
        for (int r = 0; r < 8; ++r) {
            if (acc0[r] < best0[r]) { best0[r] = acc0[r]; bidx0[r] = n0 + m; }
            if (acc1[r] < best1[r]) { best1[r] = acc1[r]; bidx1[r] = n0 + m; }
        }

        #pragma unroll
        for (int ks = 0; ks < 16; ++ks) bc[ks] = bn[ks];   // rotate (register rename)
        csc = csn;
    }

    // ---- argmin reduction across the 16 lanes of each half-wave ----
    // ties break toward the smaller index (reference first-min semantics)
    #pragma unroll
    for (int r = 0; r < 8; ++r) {
        #pragma unroll
        for (int off = 8; off >= 1; off >>= 1) {
            float ob = __shfl_xor(best0[r], off, 32);
            int   oi = __shfl_xor(bidx0[r], off, 32);
            if (ob < best0[r] || (ob == best0[r] && oi < bidx0[r])) {
                best0[r] = ob; bidx0[r] = oi;
            }
            float pb = __shfl_xor(best1[r], off, 32);
            int   pi = __shfl_xor(bidx1[r], off, 32);
            if (pb < best1[r] || (pb == best1[r] && pi < bidx1[r])) {
                best1[r] = pb; bidx1[r] = pi;
            }
        }
    }
    // chain0: lanes 0-15 -> rows r, lanes 16-31 -> rows r+8
    // chain1: same, for rows +16

    // ---- gather winning codebook rows from LDS, store 32x64 fp32 output ----
    #pragma unroll
    for (int r = 0; r < 8; ++r) {
        const int iA = __shfl(bidx0[r], 0, 32);    // row r
        const int iB = __shfl(bidx0[r], 16, 32);   // row r + 8
        const int iC = __shfl(bidx1[r], 0, 32);    // row r + 16
        const int iD = __shfl(bidx1[r], 16, 32);   // row r + 24
        v2f vA = *(const v2f*)(s_cb + iA * ROW_PAD + lane * 2);
        v2f vB = *(const v2f*)(s_cb + iB * ROW_PAD + lane * 2);
        v2f vC = *(const v2f*)(s_cb + iC * ROW_PAD + lane * 2);
        v2f vD = *(const v2f*)(s_cb + iD * ROW_PAD + lane * 2);
        *(v2f*)(out + (row0 + r)      * D + lane * 2) = vA;
        *(v2f*)(out + (row0 + r + 8)  * D + lane * 2) = vB;
        *(v2f*)(out + (row0 + r + 16) * D + lane * 2) = vC;
        *(v2f*)(out + (row0 + r + 24) * D + lane * 2) = vD;
    }
}

extern "C" void kernel_launch(void* const* d_in, const int* in_sizes, int n_in,
                              void* d_out, int out_size, void* d_ws, size_t ws_size,
                              hipStream_t stream) {
    const float* x  = (const float*)d_in[0];   // [64,32,32,64] fp32
    const float* cb = (const float*)d_in[1];   // [1024,64]     fp32
    float* out = (float*)d_out;                // [64,32,32,64] fp32

    const int N    = in_sizes[0] / D;          // 65536 query vectors
    const int grid = N / ROWS_PER_WG;          // 256 workgroups (~1 per WGP)

    const size_t smem_bytes = (size_t)(K * ROW_PAD + K) * sizeof(float); // ~270 KB

    // CDNA5 WGP has 320 KB LDS; opt in to >64 KB dynamic LDS.
    hipFuncSetAttribute((const void*)vq_wmma_f32_kernel,
                        hipFuncAttributeMaxDynamicSharedMemorySize,
                        (int)smem_bytes);

    vq_wmma_f32_kernel<<<grid, BLOCK, smem_bytes, stream>>>(x, cb, out);
}